// GAT_75952201662531
// MI455X (gfx1250) — compile-verified
//
#include <hip/hip_runtime.h>
#include <cstdint>
#include <cstddef>

// ---- problem constants (from reference) ----
#define N_NODES   50000
#define N_EDGES   800000
#define TOT_E     (N_EDGES + N_NODES)   // edges + self loops = 850000
#define IN_C      128
#define HID       32
#define OUT_C     16
#define HEADS     8
#define HIDDEN    (HEADS * HID)         // 256
#define NEG_SLOPE 0.2f
#define GAT_EPS   1e-16f

typedef float v2f __attribute__((ext_vector_type(2)));
typedef float v8f __attribute__((ext_vector_type(8)));

// ======================================================================
// FP32 WMMA GEMM: C[M,NC] = A[M,K] @ B[K,NC], K and NC compile-time.
// One wave computes a 16 x (16*TN) strip. The K loop is fully unrolled so
// every A/B access is base-pointer + immediate offset (no per-step 64-bit
// address math), and the TN fragment loads of each step are grouped ahead
// of the TN WMMAs so loads clause together and WMMAs issue back-to-back
// into TN independent accumulator chains.
// Fragment layouts per CDNA5 ISA 7.12.2:
//   A 16x4 : lanes 0-15 hold row M=lane  {v0=K0, v1=K1}; lanes 16-31 {K2,K3}
//   B 4x16 : lanes 0-15 hold col N=lane  {v0=K0, v1=K1}; lanes 16-31 {K2,K3}
//   C 16x16: VGPR r -> row r (lanes 0-15) / row r+8 (lanes 16-31), col=lane&15
// ======================================================================
template <int TN, int K, int NC>
__global__ __launch_bounds__(128) void gat_wmma_gemm_f32(
    const float* __restrict__ A, const float* __restrict__ B,
    float* __restrict__ C, int M, int waveTilesN)
{
    const int wave = blockIdx.x * (blockDim.x >> 5) + (threadIdx.x >> 5);
    const int lane = threadIdx.x & 31;
    const int tM = wave / waveTilesN;       // 16-row tile index
    const int tg = wave % waveTilesN;       // group of TN column tiles
    if (tM * 16 >= M) return;               // wave-uniform: EXEC stays all-1s

    const int half = lane >> 4;             // 0: K pair {0,1}, 1: pair {2,3}
    const int l    = lane & 15;
    // Base pointers; all loop accesses are immediate offsets from these.
    const float* __restrict__ Ap = A + (size_t)(tM * 16 + l) * K + half * 2;
    const float* __restrict__ Bp = B + (size_t)(half * 2) * NC + tg * 16 * TN + l;

    const v8f zero = {0.f,0.f,0.f,0.f,0.f,0.f,0.f,0.f};
    v8f c[TN];
#pragma unroll
    for (int j = 0; j < TN; ++j) c[j] = zero;

#pragma unroll
    for (int k = 0; k < K; k += 4) {
        const v2f a = *(const v2f*)(Ap + k);        // 8B-aligned pair {ka, ka+1}
        v2f b[TN];
#pragma unroll
        for (int j = 0; j < TN; ++j) {              // grouped loads -> one clause
            b[j].x = Bp[(size_t)k * NC + j * 16];
            b[j].y = Bp[(size_t)(k + 1) * NC + j * 16];
        }
#pragma unroll
        for (int j = 0; j < TN; ++j)                // back-to-back WMMAs
            c[j] = __builtin_amdgcn_wmma_f32_16x16x4_f32(
                       /*neg_a=*/false, a, /*neg_b=*/false, b[j],
                       /*c_mod=*/(short)0, c[j],
                       /*reuse_a=*/false, /*reuse_b=*/false);
    }

    float* __restrict__ Cbase =
        C + (size_t)(tM * 16 + half * 8) * NC + tg * 16 * TN + l;
#pragma unroll
    for (int j = 0; j < TN; ++j)
#pragma unroll
        for (int r = 0; r < 8; ++r)
            Cbase[(size_t)r * NC + j * 16] = c[j][r];
}

// ======================================================================
// Per-node attention logits: asrc[n,h] = sum_c xl[n,h,c]*att_src[h,c]
// ======================================================================
__global__ __launch_bounds__(256) void gat_att_logits(
    const float* __restrict__ xl, const float* __restrict__ att_s,
    const float* __restrict__ att_d, float* __restrict__ asrc,
    float* __restrict__ adst, int n, int H, int CH)
{
    const int i = blockIdx.x * blockDim.x + threadIdx.x;
    if (i >= n * H) return;
    const int h = i % H;
    const float* __restrict__ row = xl + (size_t)i * CH;
    const float* __restrict__ as  = att_s + (size_t)h * CH;
    const float* __restrict__ ad  = att_d + (size_t)h * CH;
    float s = 0.f, d = 0.f;
    for (int c = 0; c < CH; ++c) { float v = row[c]; s += v * as[c]; d += v * ad[c]; }
    asrc[i] = s;
    adst[i] = d;
}

// Monotonic uint key for float atomic-max; memset-0 init < every real key.
__device__ __forceinline__ unsigned fkey(float f) {
    unsigned u = __float_as_uint(f);
    return (u & 0x80000000u) ? ~u : (u | 0x80000000u);
}
__device__ __forceinline__ float fdec(unsigned k) {
    unsigned u = (k & 0x80000000u) ? (k ^ 0x80000000u) : ~k;
    return __uint_as_float(u);
}

__device__ __forceinline__ void edge_sd(long long e, const int* __restrict__ src,
                                        const int* __restrict__ dst, int& s, int& d) {
    if (e < N_EDGES) { s = src[e]; d = dst[e]; }
    else             { s = (int)(e - N_EDGES); d = s; }   // appended self loops
}

// ---- softmax phase 1: per-destination running max of leaky_relu score ----
__global__ __launch_bounds__(256) void gat_edge_max(
    const int* __restrict__ src, const int* __restrict__ dst,
    const float* __restrict__ asrc, const float* __restrict__ adst,
    unsigned* __restrict__ emaxk, int H)
{
    const long long i = (long long)blockIdx.x * blockDim.x + threadIdx.x;
    if (i >= (long long)TOT_E * H) return;
    const int h = (int)(i % H);
    const long long e = i / H;
    int s, d; edge_sd(e, src, dst, s, d);
    float v = asrc[(size_t)s * H + h] + adst[(size_t)d * H + h];
    v = v > 0.f ? v : NEG_SLOPE * v;
    atomicMax(&emaxk[(size_t)d * H + h], fkey(v));
}

// ---- softmax phase 2: exp(score - max), store per-edge, sum per-dest ----
__global__ __launch_bounds__(256) void gat_edge_exp(
    const int* __restrict__ src, const int* __restrict__ dst,
    const float* __restrict__ asrc, const float* __restrict__ adst,
    const unsigned* __restrict__ emaxk, float* __restrict__ eexp,
    float* __restrict__ denom, int H)
{
    const long long i = (long long)blockIdx.x * blockDim.x + threadIdx.x;
    if (i >= (long long)TOT_E * H) return;
    const int h = (int)(i % H);
    const long long e = i / H;
    int s, d; edge_sd(e, src, dst, s, d);
    float v = asrc[(size_t)s * H + h] + adst[(size_t)d * H + h];
    v = v > 0.f ? v : NEG_SLOPE * v;
    const float m  = fdec(emaxk[(size_t)d * H + h]);
    const float ex = expf(v - m);
    eexp[i] = ex;
    atomicAdd(&denom[(size_t)d * H + h], ex);
}

// ---- phase 3: acc[d,h,c] += alpha * xl[s,h,c]  (coalesced over c) ----
__global__ __launch_bounds__(256) void gat_edge_aggregate(
    const int* __restrict__ src, const int* __restrict__ dst,
    const float* __restrict__ eexp, const float* __restrict__ denom,
    const float* __restrict__ xl, float* __restrict__ acc, int H, int CH)
{
    const long long i = (long long)blockIdx.x * blockDim.x + threadIdx.x;
    if (i >= (long long)TOT_E * H * CH) return;
    const int c        = (int)(i % CH);
    const long long eh = i / CH;
    const int h        = (int)(eh % H);
    const long long e  = eh / H;
    int s, d; edge_sd(e, src, dst, s, d);
    const float alpha = eexp[eh] / (denom[(size_t)d * H + h] + GAT_EPS);
    const float val   = alpha * xl[((size_t)s * H + h) * CH + c];
    atomicAdd(&acc[((size_t)d * H + h) * CH + c], val);
}

// ---- layer-1 epilogue: +bias, ELU, in place (acc becomes h) ----
__global__ __launch_bounds__(256) void gat_finalize_elu(
    float* __restrict__ acc, const float* __restrict__ bias, int n, int F)
{
    const int i = blockIdx.x * blockDim.x + threadIdx.x;
    if (i >= n * F) return;
    const float v = acc[i] + bias[i % F];
    acc[i] = v > 0.f ? v : expm1f(v);
}

// ---- layer-2 epilogue: +bias into d_out (heads=1, mean == identity) ----
__global__ __launch_bounds__(256) void gat_finalize_bias(
    const float* __restrict__ acc, const float* __restrict__ bias,
    float* __restrict__ out, int n, int F)
{
    const int i = blockIdx.x * blockDim.x + threadIdx.x;
    if (i >= n * F) return;
    out[i] = acc[i] + bias[i % F];
}

// ======================================================================
extern "C" void kernel_launch(void* const* d_in, const int* in_sizes, int n_in,
                              void* d_out, int out_size, void* d_ws, size_t ws_size,
                              hipStream_t stream)
{
    (void)in_sizes; (void)n_in; (void)out_size; (void)ws_size;

    const float* x    = (const float*)d_in[0];
    const int*   ei   = (const int*)  d_in[1];   // [2,E] row-major
    const float* W1   = (const float*)d_in[2];
    const float* ats1 = (const float*)d_in[3];
    const float* atd1 = (const float*)d_in[4];
    const float* b1   = (const float*)d_in[5];
    const float* W2   = (const float*)d_in[6];
    const float* ats2 = (const float*)d_in[7];
    const float* atd2 = (const float*)d_in[8];
    const float* b2   = (const float*)d_in[9];
    const int* src = ei;
    const int* dst = ei + N_EDGES;
    float* out = (float*)d_out;

    // ---- workspace layout (floats). Zero-init region first, one memset. ----
    float*    acc1 = (float*)d_ws;                     // [N,256] -> becomes h
    float*    acc2 = acc1 + (size_t)N_NODES * HIDDEN;  // [N,16]
    unsigned* emk1 = (unsigned*)(acc2 + (size_t)N_NODES * OUT_C);  // [N,8]
    float*    den1 = (float*)(emk1 + (size_t)N_NODES * HEADS);     // [N,8]
    unsigned* emk2 = (unsigned*)(den1 + (size_t)N_NODES * HEADS);  // [N]
    float*    den2 = (float*)(emk2 + (size_t)N_NODES);             // [N]
    const size_t zero_bytes =
        (size_t)N_NODES * (HIDDEN + OUT_C + 2 * HEADS + 2) * sizeof(float);
    // non-zeroed scratch
    float* xl1 = den2 + (size_t)N_NODES;               // [N,256]
    float* as1 = xl1 + (size_t)N_NODES * HIDDEN;       // [N,8]
    float* ad1 = as1 + (size_t)N_NODES * HEADS;        // [N,8]
    float* ee1 = ad1 + (size_t)N_NODES * HEADS;        // [TOT_E,8]
    // layer-2 scratch aliases ee1 (dead after layer-1 aggregate): needs 7MB < 27MB
    float* xl2 = ee1;                                  // [N,16]
    float* as2 = xl2 + (size_t)N_NODES * OUT_C;        // [N]
    float* ad2 = as2 + (size_t)N_NODES;                // [N]
    float* ee2 = ad2 + (size_t)N_NODES;                // [TOT_E]

    hipMemsetAsync(d_ws, 0, zero_bytes, stream);

    const int TB = 256;
    auto blk = [](long long n, int tb) { return (unsigned)((n + tb - 1) / tb); };

    // ---------------- Layer 1 ----------------
    {   // GEMM1: [N,128]@[128,256]; 16x64 strip per wave (TN=4), 4 waves/block
        const int waveTilesN = HIDDEN / (16 * 4);      // = 4
        const long long waves = (long long)(N_NODES / 16) * waveTilesN;
        gat_wmma_gemm_f32<4, IN_C, HIDDEN><<<blk(waves, 4), 128, 0, stream>>>(
            x, W1, xl1, N_NODES, waveTilesN);
    }
    gat_att_logits<<<blk((long long)N_NODES * HEADS, TB), TB, 0, stream>>>(
        xl1, ats1, atd1, as1, ad1, N_NODES, HEADS, HID);
    gat_edge_max<<<blk((long long)TOT_E * HEADS, TB), TB, 0, stream>>>(
        src, dst, as1, ad1, emk1, HEADS);
    gat_edge_exp<<<blk((long long)TOT_E * HEADS, TB), TB, 0, stream>>>(
        src, dst, as1, ad1, emk1, ee1, den1, HEADS);
    gat_edge_aggregate<<<blk((long long)TOT_E * HEADS * HID, TB), TB, 0, stream>>>(
        src, dst, ee1, den1, xl1, acc1, HEADS, HID);
    gat_finalize_elu<<<blk((long long)N_NODES * HIDDEN, TB), TB, 0, stream>>>(
        acc1, b1, N_NODES, HIDDEN);

    // ---------------- Layer 2 (heads=1, ch=16) ----------------
    {   // GEMM2: [N,256]@[256,16]; single 16-col tile per wave (TN=1)
        const int waveTilesN = OUT_C / 16;             // = 1
        const long long waves = (long long)(N_NODES / 16) * waveTilesN;
        gat_wmma_gemm_f32<1, HIDDEN, OUT_C><<<blk(waves, 4), 128, 0, stream>>>(
            acc1, W2, xl2, N_NODES, waveTilesN);
    }
    gat_att_logits<<<blk((long long)N_NODES, TB), TB, 0, stream>>>(
        xl2, ats2, atd2, as2, ad2, N_NODES, 1, OUT_C);
    gat_edge_max<<<blk((long long)TOT_E, TB), TB, 0, stream>>>(
        src, dst, as2, ad2, emk2, 1);
    gat_edge_exp<<<blk((long long)TOT_E, TB), TB, 0, stream>>>(
        src, dst, as2, ad2, emk2, ee2, den2, 1);
    gat_edge_aggregate<<<blk((long long)TOT_E * OUT_C, TB), TB, 0, stream>>>(
        src, dst, ee2, den2, xl2, acc2, 1, OUT_C);
    gat_finalize_bias<<<blk((long long)N_NODES * OUT_C, TB), TB, 0, stream>>>(
        acc2, b2, out, N_NODES, OUT_C);
}